// LogSigLayer_5394478924085
// MI455X (gfx1250) — compile-verified
//
#include <hip/hip_runtime.h>

// CDNA5 / gfx1250: wave32, WMMA + async global->LDS staging.
typedef float v2f __attribute__((ext_vector_type(2)));
typedef float v4f __attribute__((ext_vector_type(4)));
typedef float v8f __attribute__((ext_vector_type(8)));

#define WAVES_PER_BLOCK 4
#define KC 16                 // time-steps per chunk
#define NCHUNK 16             // 16*16 = 256 padded steps (255 real + 1 zero pad)
#define XBUF_FLOATS 2048      // whole path x[256][8] per wave
#define WBUF_FLOATS 1024      // W-tile [16][64]
#define DXC_FLOATS  128       // dx chunk [16][8]
#define LDS_PER_WAVE (XBUF_FLOATS + WBUF_FLOATS + DXC_FLOATS)   // 3200 floats

// All LDS here is wave-private. CDNA5 LDS ops from one wave execute in order
// (DScnt), so cross-lane handoffs need only a compiler-ordering fence, not a
// workgroup barrier.
__device__ __forceinline__ void wave_sync() {
  __builtin_amdgcn_fence(__ATOMIC_ACQ_REL, "wavefront");
  __builtin_amdgcn_wave_barrier();
  __builtin_amdgcn_fence(__ATOMIC_ACQ_REL, "wavefront");
}

__global__ __launch_bounds__(WAVES_PER_BLOCK * 32)
void logsig_wmma_kernel(const float* __restrict__ x, float* __restrict__ out, int B) {
  __shared__ float lds[WAVES_PER_BLOCK * LDS_PER_WAVE];

  const int lane = threadIdx.x & 31;
  const int wv   = threadIdx.x >> 5;
  int b = blockIdx.x * WAVES_PER_BLOCK + wv;
  if (b >= B) b = B - 1;   // clamp (duplicate work writes identical values)

  float* xbuf = &lds[wv * LDS_PER_WAVE];   // [256][8] raw path
  float* Wbuf = xbuf + XBUF_FLOATS;        // [16][64] W-tile for current chunk
  float* dxc  = Wbuf + WBUF_FLOATS;        // [16][8] dx of current chunk

  const int j  = lane & 7;    // column channel owned in the scan
  const int i1 = lane >> 3;   // row channel (0..3); second row = i1+4
  const int hi = lane >> 4;   // WMMA half-wave selector
  const int n  = lane & 15;   // WMMA M-row / N-col index

  const float* xb = x + (size_t)b * 256 * 8;

  // ---- stage the whole path into LDS with CDNA5 async copies (ASYNCcnt) ----
  {
    const uint32_t lbase = (uint32_t)(uintptr_t)xbuf;   // LDS byte offset
    #pragma unroll
    for (int r = 0; r < 16; ++r) {
      const int idx = r * 32 + lane;                     // float4 index, 512 total
      const unsigned long long g = (unsigned long long)(uintptr_t)(xb + idx * 4);
      const uint32_t l = lbase + idx * 16;
      asm volatile("global_load_async_to_lds_b128 %0, %1, off"
                   :: "v"(l), "v"(g) : "memory");
    }
    asm volatile("s_wait_asynccnt 0x0" ::: "memory");
  }
  wave_sync();

  // Scan carries: p = S1 prefix (replicated per 8-lane group), q = S2 prefix.
  float p1 = 0.f, p2 = 0.f, q1 = 0.f, q2 = 0.f;
  // S3 accumulators: 4 tiles of 16 rows (ij) x 16 cols (k, 8 used).
  v8f acc0 = {}, acc1 = {}, acc2 = {}, acc3 = {};

  for (int c = 0; c < NCHUNK; ++c) {
    const int t0 = c * KC;

    // ---- dx chunk from LDS: one float4 per lane ----
    {
      const int row = t0 + (lane >> 1);          // dx row of this lane's float4
      v4f d = {};
      if (row < 255) {
        const v4f* x4 = (const v4f*)xbuf;
        d = x4[(row + 1) * 2 + (lane & 1)] - x4[row * 2 + (lane & 1)];
      }
      ((v4f*)dxc)[lane] = d;
    }
    wave_sync();

    // ---- sequential Chen scan over the chunk; emit W rows to LDS ----
    #pragma unroll
    for (int tl = 0; tl < KC; ++tl) {
      const float dxj  = dxc[tl * 8 + j];
      const float dxi1 = dxc[tl * 8 + i1];
      const float dxi2 = dxc[tl * 8 + i1 + 4];
      // W_t[ij] = Q_t[ij] + 0.5*P_t[i]*dx[j] + (1/6)*dx[i]*dx[j]
      const float w1 = q1 + 0.5f * p1 * dxj + (1.f / 6.f) * dxi1 * dxj;
      const float w2 = q2 + 0.5f * p2 * dxj + (1.f / 6.f) * dxi2 * dxj;
      Wbuf[tl * 64 + lane]      = w1;
      Wbuf[tl * 64 + lane + 32] = w2;
      // u_t[ij] = P_t[i]*dx[j] + 0.5*dx[i]*dx[j]  (S2 increment)
      q1 += p1 * dxj + 0.5f * dxi1 * dxj;
      q2 += p2 * dxj + 0.5f * dxi2 * dxj;
      p1 += dxi1;
      p2 += dxi2;
    }
    wave_sync();

    // ---- S3 += W^T (64xKC) * DX (KCx8) via WMMA f32 16x16x4 ----
    #pragma unroll
    for (int g = 0; g < 4; ++g) {
      const int kt = g * 4 + 2 * hi;          // this half-wave's K rows {kt, kt+1}
      v2f bm;                                  // B[k,n] = dx[t0+k][n], 0 for n>=8
      bm.x = (n < 8) ? dxc[kt * 8 + n]       : 0.f;
      bm.y = (n < 8) ? dxc[(kt + 1) * 8 + n] : 0.f;
      v2f a0, a1, a2, a3;                      // A[m,k] = W[t0+k][mt*16+m]
      a0.x = Wbuf[kt * 64 +  0 + n];  a0.y = Wbuf[(kt + 1) * 64 +  0 + n];
      a1.x = Wbuf[kt * 64 + 16 + n];  a1.y = Wbuf[(kt + 1) * 64 + 16 + n];
      a2.x = Wbuf[kt * 64 + 32 + n];  a2.y = Wbuf[(kt + 1) * 64 + 32 + n];
      a3.x = Wbuf[kt * 64 + 48 + n];  a3.y = Wbuf[(kt + 1) * 64 + 48 + n];
      acc0 = __builtin_amdgcn_wmma_f32_16x16x4_f32(false, a0, false, bm, (short)0, acc0, false, false);
      acc1 = __builtin_amdgcn_wmma_f32_16x16x4_f32(false, a1, false, bm, (short)0, acc1, false, false);
      acc2 = __builtin_amdgcn_wmma_f32_16x16x4_f32(false, a2, false, bm, (short)0, acc2, false, false);
      acc3 = __builtin_amdgcn_wmma_f32_16x16x4_f32(false, a3, false, bm, (short)0, acc3, false, false);
    }
    wave_sync();   // WMMA reads done before next chunk overwrites Wbuf/dxc
  }

  // ---- epilogue: log() correction in the tensor algebra ----
  float* s1buf = Wbuf;       // 8 floats   (S1)
  float* s2buf = Wbuf + 8;   // 64 floats  (S2)
  if ((lane & 7) == 0) { s1buf[i1] = p1; s1buf[i1 + 4] = p2; }
  s2buf[lane]      = q1;
  s2buf[lane + 32] = q2;
  wave_sync();

  float* ob = out + (size_t)b * 584;

  // l1 = S1
  if (lane < 8) ob[lane] = s1buf[lane];

  // l2 = S2 - 0.5 * S1 (x) S1   (lane owns ij = lane and lane+32)
  {
    const float s1j = s1buf[j];
    ob[8 + lane]      = q1 - 0.5f * p1 * s1j;
    ob[8 + lane + 32] = q2 - 0.5f * p2 * s1j;
  }

  // l3 = S3 - 0.5*(S1(x)S2 + S2(x)S1) + (1/3) S1(x)S1(x)S1
  const int k = n;           // column index in C/D layout
  if (k < 8) {
    const float s1k = s1buf[k];
    #pragma unroll
    for (int mt = 0; mt < 4; ++mt) {
      const v8f acc = (mt == 0) ? acc0 : (mt == 1) ? acc1 : (mt == 2) ? acc2 : acc3;
      const int i = 2 * mt + hi;           // ij = mt*16 + v + 8*hi  ->  i = 2mt+hi
      const float s1i = s1buf[i];
      #pragma unroll
      for (int v = 0; v < 8; ++v) {
        const int ij = i * 8 + v;          // j = v
        const float val = acc[v]
            - 0.5f * (s1i * s2buf[v * 8 + k] + s2buf[ij] * s1k)
            + (1.f / 3.f) * s1i * s1buf[v] * s1k;
        ob[72 + ij * 8 + k] = val;
      }
    }
  }
}

extern "C" void kernel_launch(void* const* d_in, const int* in_sizes, int n_in,
                              void* d_out, int out_size, void* d_ws, size_t ws_size,
                              hipStream_t stream) {
  (void)n_in; (void)out_size; (void)d_ws; (void)ws_size;
  const float* x = (const float*)d_in[0];
  float* out = (float*)d_out;
  const int B = in_sizes[0] / (256 * 8);   // [B, 256, 8] fp32 path
  const int blocks = (B + WAVES_PER_BLOCK - 1) / WAVES_PER_BLOCK;
  logsig_wmma_kernel<<<blocks, WAVES_PER_BLOCK * 32, 0, stream>>>(x, out, B);
}